// T_Closure_55903294324914
// MI455X (gfx1250) — compile-verified
//
#include <hip/hip_runtime.h>
#include <hip/hip_bf16.h>

#define DIM 128
#define EPSF 1e-16f

typedef float v2f __attribute__((ext_vector_type(2)));
typedef float v8f __attribute__((ext_vector_type(8)));

// Monotone float <-> uint mapping so we can use integer atomicMax for float max.
__device__ __forceinline__ unsigned flip_f32(float f) {
  unsigned u = __float_as_uint(f);
  return u ^ ((unsigned)((int)u >> 31) | 0x80000000u);
}
__device__ __forceinline__ float unflip_f32(unsigned u) {
  unsigned b = (u & 0x80000000u) ? (u ^ 0x80000000u) : ~u;
  return __uint_as_float(b);
}

// ---------------------------------------------------------------- init
__global__ void init_kernel(float* __restrict__ agg, unsigned* __restrict__ maxu,
                            float* __restrict__ denom, float* __restrict__ attsum, int N) {
  int i = blockIdx.x * blockDim.x + threadIdx.x;
  if (i < N * DIM) agg[i] = 0.f;
  if (i < N) { maxu[i] = 0u; denom[i] = 0.f; attsum[i] = 0.f; }
}

// ------------------------------------------------- attention raw score
// One wave (32 lanes) per edge; each lane covers 4 of the 128 features.
__global__ void att_kernel(const float* __restrict__ x, const int* __restrict__ ei,
                           const int* __restrict__ etype, const float* __restrict__ W_att,
                           const float* __restrict__ b_att, const float* __restrict__ edge_emb,
                           float* __restrict__ att, unsigned* __restrict__ maxu, int E) {
  int e = (blockIdx.x * blockDim.x + threadIdx.x) >> 5;
  int lane = threadIdx.x & 31;
  if (e >= E) return;
  int src = ei[e];
  int dst = ei[E + e];
  int et  = etype[e];
  const float* xi  = x + (size_t)dst * DIM;
  const float* xj  = x + (size_t)src * DIM;
  const float* emb = edge_emb + (size_t)et * DIM;
  float p = 0.f;
#pragma unroll
  for (int i = 0; i < 4; ++i) {
    int k = lane + 32 * i;
    float m = emb[k] * xj[k];
    p += xi[k] * W_att[k] + m * W_att[DIM + k];
  }
#pragma unroll
  for (int off = 16; off > 0; off >>= 1) p += __shfl_xor(p, off, 32);
  if (lane == 0) {
    float a = p + b_att[0];
    a = (a >= 0.f) ? a : (-0.1f * a);   // leaky_relu(a, -0.1), per reference
    att[e] = a;
    atomicMax(&maxu[dst], flip_f32(a));
  }
}

// ------------------------------------------------- exp + denom segment sum
__global__ void exp_kernel(const int* __restrict__ ei, const unsigned* __restrict__ maxu,
                           float* __restrict__ att, float* __restrict__ denom, int E) {
  int e = blockIdx.x * blockDim.x + threadIdx.x;
  if (e >= E) return;
  int dst = ei[E + e];
  float m = unflip_f32(maxu[dst]);
  float v = expf(att[e] - m);
  att[e] = v;
  atomicAdd(&denom[dst], v);
}

// ------------------------------------------------- normalized att * msg scatter
__global__ void agg_kernel(const float* __restrict__ x, const int* __restrict__ ei,
                           const int* __restrict__ etype, const float* __restrict__ edge_emb,
                           const float* __restrict__ att, const float* __restrict__ denom,
                           float* __restrict__ agg, float* __restrict__ attsum, int E) {
  int e = (blockIdx.x * blockDim.x + threadIdx.x) >> 5;
  int lane = threadIdx.x & 31;
  if (e >= E) return;
  int src = ei[e];
  int dst = ei[E + e];
  int et  = etype[e];
  float a = att[e] / (denom[dst] + EPSF);
  const float* xj  = x + (size_t)src * DIM;
  const float* emb = edge_emb + (size_t)et * DIM;
  float* row = agg + (size_t)dst * DIM;
#pragma unroll
  for (int i = 0; i < 4; ++i) {
    int k = lane + 32 * i;
    atomicAdd(&row[k], a * emb[k] * xj[k]);
  }
  if (lane == 0) atomicAdd(&attsum[dst], a);
}

// ------------------------------------------------- out = agg @ W_r + attsum*b_r
// Workgroup = 16 node rows. 8 waves, wave w computes the 16x16 tile of columns
// [16w, 16w+16). K=128 swept in 32 steps of V_WMMA_F32_16X16X4_F32.
// A fragment (16x4 f32): lanes 0-15 hold M=lane, VGPR0={K0 | K2}, VGPR1={K1 | K3}.
// B fragment (4x16 f32): mirrored layout, lanes hold N=lane&15, hi half holds K+2/K+3.
__global__ void __launch_bounds__(256) gemm_kernel(const float* __restrict__ agg,
                                                   const float* __restrict__ W_r,
                                                   const float* __restrict__ b_r,
                                                   const float* __restrict__ attsum,
                                                   float* __restrict__ out, int N) {
  __shared__ float a_lds[16 * 132];   // stride 132 -> conflict-free column reads
  __shared__ float s_attsum[16];

  int base = blockIdx.x * 16;

  // Stage the 16x128 accumulator block into LDS (8 floats per thread).
  for (int j = threadIdx.x; j < 16 * DIM; j += 256) {
    int r = j >> 7;
    int c = j & (DIM - 1);
    int n = base + r;
    a_lds[r * 132 + c] = (n < N) ? agg[(size_t)n * DIM + c] : 0.f;
  }
  if (threadIdx.x < 16) {
    int n = base + threadIdx.x;
    s_attsum[threadIdx.x] = (n < N) ? attsum[n] : 0.f;
  }
  __syncthreads();

  int lane = threadIdx.x & 31;
  int w    = threadIdx.x >> 5;        // column tile 0..7
  int hi   = lane >> 4;               // half-wave selector
  int row  = lane & 15;               // M index for A
  int col  = w * 16 + (lane & 15);    // N index for B/C/D

  v8f c = {};
#pragma unroll
  for (int k0 = 0; k0 < DIM; k0 += 4) {
    int kk = k0 + 2 * hi;
    v2f a, b;
    a.x = a_lds[row * 132 + kk];
    a.y = a_lds[row * 132 + kk + 1];
    b.x = W_r[(size_t)kk * DIM + col];
    b.y = W_r[(size_t)(kk + 1) * DIM + col];
    c = __builtin_amdgcn_wmma_f32_16x16x4_f32(false, a, false, b,
                                              (short)0, c, false, false);
  }

  float br = b_r[col];
#pragma unroll
  for (int v = 0; v < 8; ++v) {
    int m = v + 8 * hi;               // C/D layout: VGPR v holds M=v (lo) / M=v+8 (hi)
    int orow = base + m;
    if (orow < N) out[(size_t)orow * DIM + col] = c[v] + s_attsum[m] * br;
  }
}

// ---------------------------------------------------------------- launch
extern "C" void kernel_launch(void* const* d_in, const int* in_sizes, int n_in,
                              void* d_out, int out_size, void* d_ws, size_t ws_size,
                              hipStream_t stream) {
  const float* x        = (const float*)d_in[0];
  const int*   ei       = (const int*)d_in[1];   // [0..E)=src, [E..2E)=dst
  const int*   etype    = (const int*)d_in[2];
  const float* W_r      = (const float*)d_in[3];
  const float* b_r      = (const float*)d_in[4];
  const float* W_att    = (const float*)d_in[5];
  const float* b_att    = (const float*)d_in[6];
  const float* edge_emb = (const float*)d_in[7];
  float* out = (float*)d_out;

  int E = in_sizes[2];
  int N = in_sizes[0] / DIM;

  char* ws = (char*)d_ws;
  float*    att    = (float*)ws;
  unsigned* maxu   = (unsigned*)(ws + (size_t)E * 4);
  float*    denom  = (float*)(ws + (size_t)E * 4 + (size_t)N * 4);
  float*    attsum = (float*)(ws + (size_t)E * 4 + (size_t)N * 8);

  const int T = 256;
  init_kernel<<<(N * DIM + T - 1) / T, T, 0, stream>>>(out, maxu, denom, attsum, N);
  att_kernel <<<(E + 7) / 8, T, 0, stream>>>(x, ei, etype, W_att, b_att, edge_emb,
                                             att, maxu, E);
  exp_kernel <<<(E + T - 1) / T, T, 0, stream>>>(ei, maxu, att, denom, E);
  agg_kernel <<<(E + 7) / 8, T, 0, stream>>>(x, ei, etype, edge_emb, att, denom,
                                             out, attsum, E);
  gemm_kernel<<<(N + 15) / 16, T, 0, stream>>>(out, W_r, b_r, attsum, out, N);
}